// FGFI_1288490188935
// MI455X (gfx1250) — compile-verified
//
#include <hip/hip_runtime.h>

typedef float v2f __attribute__((ext_vector_type(2)));
typedef float v4f __attribute__((ext_vector_type(4)));
typedef float v8f __attribute__((ext_vector_type(8)));

// ---------------------------------------------------------------------------
// Exact-f32 full-wave (32 lane) sum using V_WMMA_F32_16X16X4_F32.
// A layout (ISA 7.12.2): lanes 0-15: V0=K0, V1=K1 ; lanes 16-31: V0=K2, V1=K3,
// row M = lane%16.  With a=(v,0) and B=ones: D[m][n] = v[m] + v[m+16] for all n.
// D vgpr j holds rows j (lanes 0-15) and j+8 (lanes 16-31), so summing the 8
// D registers gives half-wave sums; one shfl_xor(16) completes the reduction.
// All float32, round-to-nearest-even -> no precision loss vs a shuffle tree.
// ---------------------------------------------------------------------------
__device__ __forceinline__ float wave_sum_wmma(float v) {
  v2f a; a[0] = v;    a[1] = 0.0f;
  v2f b; b[0] = 1.0f; b[1] = 1.0f;
  v8f c = {};
  v8f d = __builtin_amdgcn_wmma_f32_16x16x4_f32(
      /*neg_a=*/false, a, /*neg_b=*/false, b,
      /*c_mod=*/(short)0, c, /*reuse_a=*/false, /*reuse_b=*/false);
  float s = ((d[0] + d[1]) + (d[2] + d[3])) + ((d[4] + d[5]) + (d[6] + d[7]));
  s += __shfl_xor(s, 16, 32);
  return s;  // uniform across the wave
}

// ---------------------------------------------------------------------------
// Channel loop: 64 channels, float4 per lane. NT=true streams past L2
// (level 0, 210MB, would only thrash); NT=false lets levels 1-2 (65MB) stay
// L2-resident across graph replays.
// ---------------------------------------------------------------------------
template <bool NT>
__device__ __forceinline__ float ssq_loop(const float* __restrict__ p,
                                          const float* __restrict__ t,
                                          size_t off4, int stride4, v4f mv) {
  const v4f* __restrict__ pp = (const v4f*)p + off4;
  const v4f* __restrict__ tt = (const v4f*)t + off4;
  v4f acc = {0.f, 0.f, 0.f, 0.f};
#pragma unroll 4
  for (int c = 0; c < 64; ++c) {
    v4f a = NT ? __builtin_nontemporal_load(pp) : *pp;
    v4f b = NT ? __builtin_nontemporal_load(tt) : *tt;
    v4f d = (a - b) * mv;   // mv components are 0.0 or 1.0
    acc += d * d;
    pp += stride4;
    tt += stride4;
  }
  return (acc[0] + acc[1]) + (acc[2] + acc[3]);
}

// ---------------------------------------------------------------------------
// ws layout: [0..48)  6 doubles: ssq[3], sumM[3]
//            [64..)   byte masks: lvl0 16*6400, lvl1 16*1600, lvl2 16*400
// ---------------------------------------------------------------------------
__global__ void fgfi_init(double* __restrict__ accs, unsigned* __restrict__ mw,
                          int nwords) {
  int i = blockIdx.x * blockDim.x + threadIdx.x;
  if (i < 6) accs[i] = 0.0;
  for (int j = i; j < nwords; j += gridDim.x * blockDim.x) mw[j] = 0u;
}

// One block per box. Identical racy byte writes of 1 are benign.
__global__ void fgfi_mask(const float* __restrict__ bboxes,
                          const int* __restrict__ batch_idx,
                          unsigned char* __restrict__ mask, int S) {
  int tb = blockIdx.x;
  float fS = (float)S;
  int xc = (int)floorf(bboxes[tb * 4 + 0] * fS);
  int yc = (int)floorf(bboxes[tb * 4 + 1] * fS);
  int w  = (int)floorf(bboxes[tb * 4 + 2] * fS);
  int h  = (int)floorf(bboxes[tb * 4 + 3] * fS);
  int xl = max(xc - w / 2, 0);
  int yt = max(yc - h / 2, 0);
  int xr = min(xc + w / 2, S - 1);
  int yd = min(yc + h / 2, S - 1);
  int W = xr - xl + 1;
  int area = W * (yd - yt + 1);
  unsigned char* mb = mask + (size_t)batch_idx[tb] * S * S;
  for (int i = threadIdx.x; i < area; i += blockDim.x) {
    int y = yt + i / W, x = xl + i % W;
    mb[y * S + x] = (unsigned char)1;
  }
}

// Fused 3-level masked sum-of-squares + mask count.
// 1 wave = 1 (chunk of 128 positions) x (64-channel group); 4 groups/chunk.
// grid = (400, 3); levels 1/2 retire extra blocks with a wave-uniform exit.
__global__ __launch_bounds__(256) void fgfi_reduce(
    const float* __restrict__ p0, const float* __restrict__ t0,
    const unsigned char* __restrict__ k0, const float* __restrict__ p1,
    const float* __restrict__ t1, const unsigned char* __restrict__ k1,
    const float* __restrict__ p2, const float* __restrict__ t2,
    const unsigned char* __restrict__ k2, double* __restrict__ accs) {
  const float* p;
  const float* t;
  const unsigned char* mk;
  int SS, chunks;
  const int lvl = blockIdx.y;
  if (lvl == 0)      { p = p0; t = t0; mk = k0; SS = 6400; chunks = 800; }
  else if (lvl == 1) { p = p1; t = t1; mk = k1; SS = 1600; chunks = 200; }
  else               { p = p2; t = t2; mk = k2; SS = 400;  chunks = 50;  }

  const int wave = (int)(blockIdx.x * 8u) + (int)(threadIdx.x >> 5);
  const int lane = threadIdx.x & 31;
  const int chunk = wave >> 2;   // 128 positions per wave
  const int cg = wave & 3;       // channel group (64 channels)
  if (chunk >= chunks) return;   // wave-uniform: EXEC stays all-1s below

  const int base = chunk * 128 + lane * 4;  // flat (b,yx), 4 positions/lane
  const int b = base / SS;
  const int yx = base - b * SS;

  const unsigned mw = *(const unsigned*)(mk + base);  // 4 mask bytes
  v4f mv = {(float)(mw & 0xffu), (float)((mw >> 8) & 0xffu),
            (float)((mw >> 16) & 0xffu), (float)(mw >> 24)};

  float ssq = 0.0f;
  if (__ballot(mw != 0u)) {  // skip 64KB+ of traffic for empty chunks
    size_t off4 = (size_t)(b * 256 + cg * 64) * (size_t)(SS >> 2) +
                  (size_t)(yx >> 2);
    ssq = (lvl == 0) ? ssq_loop<true>(p, t, off4, SS >> 2, mv)
                     : ssq_loop<false>(p, t, off4, SS >> 2, mv);
  }
  float msum = (cg == 0) ? (mv[0] + mv[1] + mv[2] + mv[3]) : 0.0f;

  float ssq_tot = wave_sum_wmma(ssq);
  float m_tot   = wave_sum_wmma(msum);

  if (lane == 0) {
    __hip_atomic_fetch_add(&accs[lvl], (double)ssq_tot, __ATOMIC_RELAXED,
                           __HIP_MEMORY_SCOPE_AGENT);
    if (cg == 0)
      __hip_atomic_fetch_add(&accs[3 + lvl], (double)m_tot, __ATOMIC_RELAXED,
                             __HIP_MEMORY_SCOPE_AGENT);
  }
}

__global__ void fgfi_finalize(const double* __restrict__ accs,
                              float* __restrict__ out) {
  if (threadIdx.x == 0 && blockIdx.x == 0) {
    double L = 0.0;
    for (int l = 0; l < 3; ++l) L += accs[l] / (256.0 * accs[3 + l]);
    out[0] = (float)(L / 3.0);
  }
}

extern "C" void kernel_launch(void* const* d_in, const int* in_sizes, int n_in,
                              void* d_out, int out_size, void* d_ws,
                              size_t ws_size, hipStream_t stream) {
  (void)in_sizes; (void)n_in; (void)out_size; (void)ws_size;
  // setup_inputs() dict order: pred0,true0,pred1,true1,pred2,true2,bboxes,batch_idx,cls
  const float* pred0 = (const float*)d_in[0];
  const float* true0 = (const float*)d_in[1];
  const float* pred1 = (const float*)d_in[2];
  const float* true1 = (const float*)d_in[3];
  const float* pred2 = (const float*)d_in[4];
  const float* true2 = (const float*)d_in[5];
  const float* bboxes = (const float*)d_in[6];
  const int* batch_idx = (const int*)d_in[7];

  double* accs = (double*)d_ws;
  unsigned char* m0 = (unsigned char*)d_ws + 64;
  unsigned char* m1 = m0 + 16 * 80 * 80;   // 102400 B
  unsigned char* m2 = m1 + 16 * 40 * 40;   // +25600 B (total 134400 B masks)

  fgfi_init<<<132, 256, 0, stream>>>(accs, (unsigned*)m0, 134400 / 4);
  fgfi_mask<<<128, 128, 0, stream>>>(bboxes, batch_idx, m0, 80);
  fgfi_mask<<<128, 128, 0, stream>>>(bboxes, batch_idx, m1, 40);
  fgfi_mask<<<128, 128, 0, stream>>>(bboxes, batch_idx, m2, 20);
  fgfi_reduce<<<dim3(400, 3, 1), 256, 0, stream>>>(
      pred0, true0, m0, pred1, true1, m1, pred2, true2, m2, accs);
  fgfi_finalize<<<1, 1, 0, stream>>>(accs, (float*)d_out);
}